// GNN1_49933289783570
// MI455X (gfx1250) — compile-verified
//
#include <hip/hip_runtime.h>
#include <hip/hip_bf16.h>
#include <math.h>

// ---------------------------------------------------------------------------
// Problem constants (from the reference)
// ---------------------------------------------------------------------------
#define N_NODES   50000
#define N_EDGES   800000
#define NFEAT     128
#define DIM       256
#define NCLASS    10
#define NGRAPH    128

typedef __attribute__((ext_vector_type(2))) float v2f;
typedef __attribute__((ext_vector_type(8))) float v8f;

// ---------------------------------------------------------------------------
// zero-fill (float4 granular; all our buffers are multiples of 4 floats)
// ---------------------------------------------------------------------------
__global__ void zero_kernel(float4* __restrict__ p, long long n4) {
    long long tid = (long long)blockIdx.x * blockDim.x + threadIdx.x;
    if (tid < n4) p[tid] = make_float4(0.f, 0.f, 0.f, 0.f);
}

// ---------------------------------------------------------------------------
// Edge gather * weight -> scatter-add (message passing aggregation)
// One thread handles a 4-float feature chunk of one edge.
// ---------------------------------------------------------------------------
__global__ void edge_scatter_kernel(const float* __restrict__ feat,   // [N,K]
                                    const int*  __restrict__ src,     // [E]
                                    const int*  __restrict__ dst,     // [E]
                                    const float* __restrict__ ew,     // [E]
                                    float* __restrict__ aggr,         // [N,K]
                                    long long total_chunks, int K) {
    long long tid = (long long)blockIdx.x * blockDim.x + threadIdx.x;
    if (tid >= total_chunks) return;
    const int cpe = K >> 2;                    // chunks per edge
    const long long e = tid / cpe;
    const int c = (int)(tid - e * cpe);
    const int s = src[e];
    const int d = dst[e];
    const float w = ew[e];
    const float4 v = *(const float4*)(feat + (size_t)s * K + c * 4);
    float* base = aggr + (size_t)d * K + c * 4;
    __hip_atomic_fetch_add(base + 0, v.x * w, __ATOMIC_RELAXED, __HIP_MEMORY_SCOPE_AGENT);
    __hip_atomic_fetch_add(base + 1, v.y * w, __ATOMIC_RELAXED, __HIP_MEMORY_SCOPE_AGENT);
    __hip_atomic_fetch_add(base + 2, v.z * w, __ATOMIC_RELAXED, __HIP_MEMORY_SCOPE_AGENT);
    __hip_atomic_fetch_add(base + 3, v.w * w, __ATOMIC_RELAXED, __HIP_MEMORY_SCOPE_AGENT);
}

// ---------------------------------------------------------------------------
// Fused GraphConv GEMM with f32 WMMA:
//   out[N,256] = relu( Arel @ Wrel^T + bias + Aroot @ Wroot^T )
// Block: 256 threads (8 waves). One 16-row node tile per block; each wave
// owns 2 of the 16 column tiles. A tiles staged in padded LDS.
// ---------------------------------------------------------------------------
template<int K>
__global__ __launch_bounds__(256) void node_gemm_kernel(
    const float* __restrict__ Arel,   // [N,K] aggregated messages
    const float* __restrict__ Aroot,  // [N,K] node features
    const float* __restrict__ Wrel,   // [DIM,K]
    const float* __restrict__ Wroot,  // [DIM,K]
    const float* __restrict__ bias,   // [DIM]
    float* __restrict__ out)          // [N,DIM]
{
    constexpr int LDK = K + 4;                 // pad: conflict-free ds_load_b64
    __shared__ float sA[16 * LDK];
    __shared__ float sX[16 * LDK];

    const int row0 = blockIdx.x * 16;

    // cooperative staging: 16 rows x K floats of each matrix, float4 moves
    for (int idx = threadIdx.x; idx < 16 * (K / 4); idx += 256) {
        const int r  = idx / (K / 4);
        const int c4 = (idx - r * (K / 4)) * 4;
        const float4 a = *(const float4*)(Arel  + (size_t)(row0 + r) * K + c4);
        const float4 x = *(const float4*)(Aroot + (size_t)(row0 + r) * K + c4);
        *(float4*)(sA + r * LDK + c4) = a;
        *(float4*)(sX + r * LDK + c4) = x;
    }
    __syncthreads();

    const int lane = threadIdx.x & 31;
    const int wave = threadIdx.x >> 5;
    const int mn   = lane & 15;                // M (A frag) / N (B,C frags)
    const int ksel = (lane >> 4) * 2;          // K sub-slot within a K=4 group
    const int msel = (lane >> 4) * 8;          // C-frag row offset

    const int n0a = wave * 16;                 // first column tile
    const int n0b = (wave + 8) * 16;           // second column tile

    v8f c0, c1;
    const float bv0 = bias[n0a + mn];
    const float bv1 = bias[n0b + mn];
#pragma unroll
    for (int i = 0; i < 8; ++i) { c0[i] = bv0; c1[i] = bv1; }

    // B fragment source rows: B[k][n] = W[n][k]  (C = A @ W^T)
    const float* wra = Wrel  + (size_t)(n0a + mn) * K + ksel;
    const float* wrb = Wrel  + (size_t)(n0b + mn) * K + ksel;
    const float* woa = Wroot + (size_t)(n0a + mn) * K + ksel;
    const float* wob = Wroot + (size_t)(n0b + mn) * K + ksel;

    const float* ar = sA + mn * LDK + ksel;
    const float* xr = sX + mn * LDK + ksel;

    for (int k = 0; k < K; k += 4) {
        v2f a, b0, b1;
        a.x  = ar[k];      a.y  = ar[k + 1];
        b0.x = wra[k];     b0.y = wra[k + 1];
        b1.x = wrb[k];     b1.y = wrb[k + 1];
        c0 = __builtin_amdgcn_wmma_f32_16x16x4_f32(false, a, false, b0, (short)0, c0, false, false);
        c1 = __builtin_amdgcn_wmma_f32_16x16x4_f32(false, a, false, b1, (short)0, c1, false, false);
    }
    for (int k = 0; k < K; k += 4) {
        v2f a, b0, b1;
        a.x  = xr[k];      a.y  = xr[k + 1];
        b0.x = woa[k];     b0.y = woa[k + 1];
        b1.x = wob[k];     b1.y = wob[k + 1];
        c0 = __builtin_amdgcn_wmma_f32_16x16x4_f32(false, a, false, b0, (short)0, c0, false, false);
        c1 = __builtin_amdgcn_wmma_f32_16x16x4_f32(false, a, false, b1, (short)0, c1, false, false);
    }

    // relu + store. C layout: VGPR r -> M = r + msel, N = mn
#pragma unroll
    for (int r = 0; r < 8; ++r) {
        const int m = r + msel;
        float va = c0[r]; va = va > 0.f ? va : 0.f;
        float vb = c1[r]; vb = vb > 0.f ? vb : 0.f;
        out[(size_t)(row0 + m) * DIM + n0a + mn] = va;
        out[(size_t)(row0 + m) * DIM + n0b + mn] = vb;
    }
}

// ---------------------------------------------------------------------------
// global_add_pool: g[batch[i], d] += h[i, d]
// ---------------------------------------------------------------------------
__global__ void pool_kernel(const float* __restrict__ h,   // [N,DIM]
                            const int*  __restrict__ batch,
                            float* __restrict__ g) {       // [NGRAPH,DIM]
    long long tid = (long long)blockIdx.x * blockDim.x + threadIdx.x;
    if (tid >= (long long)N_NODES * DIM) return;
    const int node = (int)(tid >> 8);
    const int d    = (int)(tid & (DIM - 1));
    const int b    = batch[node];
    __hip_atomic_fetch_add(g + (size_t)b * DIM + d, h[tid],
                           __ATOMIC_RELAXED, __HIP_MEMORY_SCOPE_AGENT);
}

// ---------------------------------------------------------------------------
// g2 = relu(g @ W_lin1^T + b)  — one block per graph
// ---------------------------------------------------------------------------
__global__ __launch_bounds__(256) void mlp1_kernel(const float* __restrict__ g,
                                                   const float* __restrict__ W,  // [DIM,DIM]
                                                   const float* __restrict__ b,
                                                   float* __restrict__ out) {
    __shared__ float srow[DIM];
    const int gi = blockIdx.x;
    const int d  = threadIdx.x;
    srow[d] = g[(size_t)gi * DIM + d];
    __syncthreads();
    float acc = b[d];
    const float* w = W + (size_t)d * DIM;
#pragma unroll 8
    for (int k = 0; k < DIM; ++k) acc += srow[k] * w[k];
    out[(size_t)gi * DIM + d] = acc > 0.f ? acc : 0.f;
}

// ---------------------------------------------------------------------------
// logits = sigmoid(g2 @ W_lin2^T + b2)
// ---------------------------------------------------------------------------
__global__ void mlp2_kernel(const float* __restrict__ g2,
                            const float* __restrict__ W,   // [NCLASS,DIM]
                            const float* __restrict__ b,
                            float* __restrict__ out) {     // [NGRAPH,NCLASS]
    const int tid = blockIdx.x * blockDim.x + threadIdx.x;
    if (tid >= NGRAPH * NCLASS) return;
    const int gi = tid / NCLASS;
    const int c  = tid - gi * NCLASS;
    float acc = b[c];
    const float* w   = W  + (size_t)c * DIM;
    const float* row = g2 + (size_t)gi * DIM;
#pragma unroll 8
    for (int k = 0; k < DIM; ++k) acc += row[k] * w[k];
    out[tid] = 1.f / (1.f + __expf(-acc));
}

// ---------------------------------------------------------------------------
// launch
// ---------------------------------------------------------------------------
extern "C" void kernel_launch(void* const* d_in, const int* in_sizes, int n_in,
                              void* d_out, int out_size, void* d_ws, size_t ws_size,
                              hipStream_t stream) {
    const float* x       = (const float*)d_in[0];
    const int*   eidx    = (const int*)  d_in[1];
    const int*   batch   = (const int*)  d_in[2];
    const float* ew      = (const float*)d_in[3];
    const float* W1_rel  = (const float*)d_in[4];
    const float* b1_rel  = (const float*)d_in[5];
    const float* W1_root = (const float*)d_in[6];
    const float* W2_rel  = (const float*)d_in[7];
    const float* b2_rel  = (const float*)d_in[8];
    const float* W2_root = (const float*)d_in[9];
    const float* W_lin1  = (const float*)d_in[10];
    const float* b_lin1  = (const float*)d_in[11];
    const float* W_lin2  = (const float*)d_in[12];
    const float* b_lin2  = (const float*)d_in[13];
    float* out = (float*)d_out;

    const int* src = eidx;            // edge_index[0]
    const int* dst = eidx + N_EDGES;  // edge_index[1]

    // workspace layout (floats)
    float* ws = (float*)d_ws;
    float* aggr1 = ws;                                  // N*NFEAT
    float* h1    = aggr1 + (size_t)N_NODES * NFEAT;     // N*DIM
    float* aggr2 = h1    + (size_t)N_NODES * DIM;       // N*DIM
    float* h2    = aggr2 + (size_t)N_NODES * DIM;       // N*DIM
    float* g     = h2    + (size_t)N_NODES * DIM;       // NGRAPH*DIM
    float* g2    = g     + (size_t)NGRAPH * DIM;        // NGRAPH*DIM

    const int TB = 256;
    auto blocks = [](long long n, int tb) { return (unsigned)((n + tb - 1) / tb); };

    // ---- layer 1 ----
    {
        long long n4 = (long long)N_NODES * NFEAT / 4;
        zero_kernel<<<blocks(n4, TB), TB, 0, stream>>>((float4*)aggr1, n4);
        long long chunks = (long long)N_EDGES * (NFEAT / 4);
        edge_scatter_kernel<<<blocks(chunks, TB), TB, 0, stream>>>(
            x, src, dst, ew, aggr1, chunks, NFEAT);
        node_gemm_kernel<NFEAT><<<N_NODES / 16, TB, 0, stream>>>(
            aggr1, x, W1_rel, W1_root, b1_rel, h1);
    }
    // ---- layer 2 ----
    {
        long long n4 = (long long)N_NODES * DIM / 4;
        zero_kernel<<<blocks(n4, TB), TB, 0, stream>>>((float4*)aggr2, n4);
        long long chunks = (long long)N_EDGES * (DIM / 4);
        edge_scatter_kernel<<<blocks(chunks, TB), TB, 0, stream>>>(
            h1, src, dst, ew, aggr2, chunks, DIM);
        node_gemm_kernel<DIM><<<N_NODES / 16, TB, 0, stream>>>(
            aggr2, h1, W2_rel, W2_root, b2_rel, h2);
    }
    // ---- pooling + MLP head ----
    {
        long long ng4 = (long long)NGRAPH * DIM / 4;
        zero_kernel<<<blocks(ng4, TB), TB, 0, stream>>>((float4*)g, ng4);
        long long nt = (long long)N_NODES * DIM;
        pool_kernel<<<blocks(nt, TB), TB, 0, stream>>>(h2, batch, g);
        mlp1_kernel<<<NGRAPH, DIM, 0, stream>>>(g, W_lin1, b_lin1, g2);
        mlp2_kernel<<<blocks(NGRAPH * NCLASS, TB), TB, 0, stream>>>(g2, W_lin2, b_lin2, out);
    }
}